// GATLayer_7009386627243
// MI455X (gfx1250) — compile-verified
//
#include <hip/hip_runtime.h>
#include <hip/hip_bf16.h>

typedef __attribute__((ext_vector_type(2))) float v2f;
typedef __attribute__((ext_vector_type(8))) float v8f;

#define IN_F   128
#define OUT_F  128
#define EDGE_D 16
#define BB     16
#define SS     1024
#define EE     8192
#define NN     (BB * SS)          // 16384 nodes
#define BE     (BB * EE)          // 131072 base edges
#define MM     (2 * BE + NN)      // 278528 directed edges incl. self loops
#define NEG_SLOPE 0.2f

// ---------------------------------------------------------------------------
// init: zero output + denom, init segmax keys to 0 (== -inf in monotonic key)
// ---------------------------------------------------------------------------
__global__ void gat_init(float* __restrict__ out, float* __restrict__ denom,
                         unsigned* __restrict__ segmax) {
  int i = blockIdx.x * blockDim.x + threadIdx.x;
  if (i < NN * OUT_F) out[i] = 0.0f;
  if (i < NN) { denom[i] = 0.0f; segmax[i] = 0u; }
}

// ---------------------------------------------------------------------------
// xt = x @ W  via V_WMMA_F32_16X16X4_F32, one wave per 16x16 output tile.
// A (16x4): lanes 0-15 -> K=k+0 (v0), K=k+1 (v1); lanes 16-31 -> K=k+2,k+3.
// B (4x16): mirrored; C/D: 8 VGPRs, lanes 0-15 -> M=r, lanes 16-31 -> M=r+8.
// ---------------------------------------------------------------------------
__global__ __launch_bounds__(256) void gat_gemm_xW(const float* __restrict__ x,
                                                   const float* __restrict__ W,
                                                   float* __restrict__ xt) {
  const int waveId  = blockIdx.x * 8 + (threadIdx.x >> 5);
  const int colTiles = OUT_F / 16;                 // 8
  const int rowBase = (waveId / colTiles) * 16;
  const int colBase = (waveId % colTiles) * 16;
  const int lane = threadIdx.x & 31;
  const int hi   = lane >> 4;                      // 0: lanes 0-15, 1: 16-31
  const int l15  = lane & 15;

  v8f c = {};
#pragma unroll
  for (int k = 0; k < IN_F; k += 4) {
    const int ka = k + hi * 2;
    v2f a, b;
    a.x = x[(rowBase + l15) * IN_F + ka];
    a.y = x[(rowBase + l15) * IN_F + ka + 1];
    b.x = W[(ka)     * OUT_F + colBase + l15];
    b.y = W[(ka + 1) * OUT_F + colBase + l15];
    c = __builtin_amdgcn_wmma_f32_16x16x4_f32(
        /*neg_a=*/false, a, /*neg_b=*/false, b,
        /*c_mod=*/(short)0, c, /*reuse_a=*/false, /*reuse_b=*/false);
  }
#pragma unroll
  for (int r = 0; r < 8; ++r)
    xt[(rowBase + r + hi * 8) * OUT_F + colBase + l15] = c[r];
}

// ---------------------------------------------------------------------------
// Per-node attention scalars: s1 = xt . a[0:128], s2 = xt . a[128:256]
// ---------------------------------------------------------------------------
__global__ void gat_node_scores(const float* __restrict__ xt,
                                const float* __restrict__ a,
                                float* __restrict__ s1, float* __restrict__ s2) {
  int n = blockIdx.x * blockDim.x + threadIdx.x;
  if (n >= NN) return;
  float acc1 = 0.0f, acc2 = 0.0f;
  const float* row = xt + (long)n * OUT_F;
#pragma unroll 8
  for (int j = 0; j < OUT_F; ++j) {
    float v = row[j];
    acc1 = fmaf(v, a[j], acc1);
    acc2 = fmaf(v, a[OUT_F + j], acc2);
  }
  s1[n] = acc1;
  s2[n] = acc2;
}

// ---------------------------------------------------------------------------
// Directed-edge decode: [0,BE) forward, [BE,2BE) reverse, [2BE,M) self loops
// ---------------------------------------------------------------------------
__device__ __forceinline__ void edge_decode(int m, const int* __restrict__ ei,
                                            int& src, int& dst, int& eaIdx) {
  if (m < 2 * BE) {
    int f = (m < BE) ? m : (m - BE);
    int b = f / EE;
    int u = ei[2 * f + 0] + b * SS;
    int v = ei[2 * f + 1] + b * SS;
    if (m < BE) { src = u; dst = v; } else { src = v; dst = u; }
    eaIdx = f;
  } else {
    int n = m - 2 * BE;
    src = n; dst = n; eaIdx = -1;
  }
}

__device__ __forceinline__ unsigned f32_to_key(float f) {
  unsigned b = __float_as_uint(f);
  return (b & 0x80000000u) ? ~b : (b | 0x80000000u);
}
__device__ __forceinline__ float key_to_f32(unsigned k) {
  unsigned b = (k & 0x80000000u) ? (k ^ 0x80000000u) : ~k;
  return __uint_as_float(b);
}

// ---------------------------------------------------------------------------
// Per-edge logit + segment max (ordered-uint atomicMax)
// ---------------------------------------------------------------------------
__global__ void gat_edge_logits(const int* __restrict__ ei,
                                const float* __restrict__ ea,
                                const float* __restrict__ a,
                                const float* __restrict__ s1,
                                const float* __restrict__ s2,
                                float* __restrict__ elog,
                                unsigned* __restrict__ segmax) {
  int m = blockIdx.x * blockDim.x + threadIdx.x;
  if (m >= MM) return;
  int src, dst, eaIdx;
  edge_decode(m, ei, src, dst, eaIdx);
  float f = s1[src] + s2[dst];
  if (eaIdx >= 0) {
    const float* e = ea + (long)eaIdx * EDGE_D;
#pragma unroll
    for (int j = 0; j < EDGE_D; ++j)
      f = fmaf(e[j], a[2 * OUT_F + j], f);
  }
  f = (f > 0.0f) ? f : NEG_SLOPE * f;   // T == 1.0
  elog[m] = f;
  atomicMax(&segmax[dst], f32_to_key(f));
}

// ---------------------------------------------------------------------------
// ex = exp(e - segmax[dst]); denom[dst] += ex
// ---------------------------------------------------------------------------
__global__ void gat_edge_exp(const int* __restrict__ ei,
                             float* __restrict__ elog,
                             const unsigned* __restrict__ segmax,
                             float* __restrict__ denom) {
  int m = blockIdx.x * blockDim.x + threadIdx.x;
  if (m >= MM) return;
  int src, dst, eaIdx;
  edge_decode(m, ei, src, dst, eaIdx);
  float ex = __expf(elog[m] - key_to_f32(segmax[dst]));
  elog[m] = ex;
  atomicAdd(&denom[dst], ex);
}

// ---------------------------------------------------------------------------
// out[dst] += (ex/denom[dst]) * xt[src]; one wave per edge, float4 per lane
// ---------------------------------------------------------------------------
__global__ __launch_bounds__(256) void gat_aggregate(const int* __restrict__ ei,
                                                     const float* __restrict__ xt,
                                                     const float* __restrict__ elog,
                                                     const float* __restrict__ denom,
                                                     float* __restrict__ out) {
  int m = blockIdx.x * 8 + (threadIdx.x >> 5);
  if (m >= MM) return;
  int lane = threadIdx.x & 31;
  int src, dst, eaIdx;
  edge_decode(m, ei, src, dst, eaIdx);
  float alpha = elog[m] / denom[dst];
  const float4* h = (const float4*)(xt + (long)src * OUT_F);
  float4 v = h[lane];
  float* o = out + (long)dst * OUT_F + lane * 4;
  atomicAdd(&o[0], alpha * v.x);
  atomicAdd(&o[1], alpha * v.y);
  atomicAdd(&o[2], alpha * v.z);
  atomicAdd(&o[3], alpha * v.w);
}

// ---------------------------------------------------------------------------
extern "C" void kernel_launch(void* const* d_in, const int* in_sizes, int n_in,
                              void* d_out, int out_size, void* d_ws, size_t ws_size,
                              hipStream_t stream) {
  const float* x  = (const float*)d_in[0];   // [B,S,128]
  const int*   ei = (const int*)d_in[1];     // [B,E,2]
  const float* ea = (const float*)d_in[2];   // [B,E,16]
  // d_in[3] node_mask, d_in[4] edge_mask: all-valid, ignored (matches reference)
  const float* W  = (const float*)d_in[5];   // [128,128]
  const float* a  = (const float*)d_in[6];   // [272,1]
  float* out = (float*)d_out;                // [B,S,128]

  // workspace carve-up (f32): xt | s1 | s2 | denom | segmax | elog
  float*    xt     = (float*)d_ws;                 // N*128
  float*    s1     = xt + (size_t)NN * OUT_F;      // N
  float*    s2     = s1 + NN;                      // N
  float*    denom  = s2 + NN;                      // N
  unsigned* segmax = (unsigned*)(denom + NN);      // N
  float*    elog   = (float*)(segmax + NN);        // M

  gat_init<<<(NN * OUT_F + 255) / 256, 256, 0, stream>>>(out, denom, segmax);

  // 1024 row-tiles * 8 col-tiles = 8192 waves, 8 waves (256 thr) per block
  gat_gemm_xW<<<(NN / 16) * (OUT_F / 16) / 8, 256, 0, stream>>>(x, W, xt);

  gat_node_scores<<<(NN + 255) / 256, 256, 0, stream>>>(xt, a, s1, s2);

  gat_edge_logits<<<(MM + 255) / 256, 256, 0, stream>>>(ei, ea, a, s1, s2, elog, segmax);

  gat_edge_exp<<<(MM + 255) / 256, 256, 0, stream>>>(ei, elog, segmax, denom);

  gat_aggregate<<<(MM + 7) / 8, 256, 0, stream>>>(ei, xt, elog, denom, out);
}